// DHPF_12756052869600
// MI455X (gfx1250) — compile-verified
//
#include <hip/hip_runtime.h>
#include <math.h>

// ---------------------------------------------------------------------------
// DHPF (FFT high-pass filter) for MI455X / gfx1250.
// 512-pt FFT = radix-16 (WMMA f32 16x16x4 chains) x radix-16 x radix-2.
// Memory-bound (~1.9 GB traffic @ 23.3 TB/s); matrix pipe does the DFT
// matmuls in full f32 precision, VALU/TRANS does twiddles, async-LDS engine
// and TDM do the global<->LDS tiling.
// ---------------------------------------------------------------------------

#define PI_F 3.14159265358979323846f

typedef __attribute__((ext_vector_type(2))) float v2f;
typedef __attribute__((ext_vector_type(8))) float v8f;

typedef __attribute__((ext_vector_type(4))) unsigned u32x4;
typedef __attribute__((ext_vector_type(8))) int i32x8;
typedef __attribute__((ext_vector_type(4))) int i32x4;

#if __has_builtin(__builtin_amdgcn_tensor_load_to_lds)
#define USE_TDM 1
#endif

// ---------------------------------------------------------------------------
// Async / TDM helpers.
// Generic pointers to LDS carry the LDS byte offset in their low 32 bits.
// ---------------------------------------------------------------------------
__device__ __forceinline__ unsigned lds_off(const void* p) {
  return (unsigned)(unsigned long long)p;
}
__device__ __forceinline__ void async_g2l_b64(void* lds_dst, const void* gsrc) {
  asm volatile("global_load_async_to_lds_b64 %0, %1, off"
               :: "v"(lds_off(lds_dst)), "v"(gsrc) : "memory");
}
__device__ __forceinline__ void async_l2g_b64(void* gdst, const void* lds_src) {
  asm volatile("global_store_async_from_lds_b64 %0, %1, off"
               :: "v"(gdst), "v"(lds_off(lds_src)) : "memory");
}
__device__ __forceinline__ void wait_async0() {
  asm volatile("s_wait_asynccnt 0x0" ::: "memory");
}
__device__ __forceinline__ void wait_ds0() {
  asm volatile("s_wait_dscnt 0x0" ::: "memory");
}

#ifdef USE_TDM
// One 1D TDM descriptor: copy nelem 8-byte elements from gptr to LDS.
__device__ __forceinline__ void tdm_load_1d(unsigned lds_addr, const void* gptr,
                                            unsigned nelem) {
  const unsigned long long ga = (unsigned long long)gptr;
  u32x4 g0;
  g0[0] = 1u;                                          // count=1, user descriptor
  g0[1] = lds_addr;                                    // lds_addr [63:32]
  g0[2] = (unsigned)(ga & 0xFFFFFFFFu);                // global_addr [95:64]
  g0[3] = (unsigned)((ga >> 32) & 0x1FFFFFFu)          // global_addr [120:96]
          | (2u << 30);                                // type=2 (image) [127:126]
  i32x8 g1;
  g1[0] = (int)(3u << 16);                             // data_size=8B [17:16]
  g1[1] = (int)((nelem & 0xFFFFu) << 16);              // tensor_dim0[15:0] @ [63:48]
  g1[2] = (int)(((nelem >> 16) & 0xFFFFu)              // tensor_dim0[31:16]
          | (1u << 16));                               // tensor_dim1=1 @ [111:80]
  g1[3] = (int)((nelem & 0xFFFFu) << 16);              // tile_dim0 @ [127:112]
  g1[4] = 0;                                           // tile_dim1=0, tile_dim2=0
  g1[5] = (int)nelem;                                  // tensor_dim0_stride[31:0]
  g1[6] = 0;
  g1[7] = 0;
  i32x4 z4 = {};
#if __has_include(<hip/amd_detail/amd_gfx1250_TDM.h>)
  i32x8 z8 = {};
  __builtin_amdgcn_tensor_load_to_lds(g0, g1, z4, z4, z8, 0);   // therock 6-arg
#else
  __builtin_amdgcn_tensor_load_to_lds(g0, g1, z4, z4, 0);        // ROCm 7.2 5-arg
#endif
}
#endif

// ---------------------------------------------------------------------------
// D = A(16x4 f32) * B(4x16 f32) + C  — chained over K for 16x16x16 matmuls.
#define WMMA_F32X4(A, B, C) \
  __builtin_amdgcn_wmma_f32_16x16x4_f32(false, (A), false, (B), (short)0, (C), false, false)

// W16 DFT matrix held in WMMA A-operand layout.
// A layout (16x4 per chunk): lanes 0-15 -> M=lane, v0=K0,v1=K1; lanes 16-31 -> M=lane-16, v0=K2,v1=K3.
// F32 WMMA has no A/B NEG modifier, so keep a pre-negated imaginary part.
struct W16 {
  v2f re[4];
  v2f im[4];
  v2f nim[4];
};

__device__ __forceinline__ W16 make_w16(float sgn) {
  W16 w;
  const int lane = threadIdx.x & 31;
  const int m = lane & 15;
  const int hb = (lane >> 4) * 2;  // K offset within 4-chunk for upper half-wave
#pragma unroll
  for (int q = 0; q < 4; ++q) {
    const float k0 = (float)(m * (4 * q + hb));
    const float k1 = (float)(m * (4 * q + hb + 1));
    float s0, c0, s1, c1;
    __sincosf(sgn * (2.0f * PI_F / 16.0f) * k0, &s0, &c0);
    __sincosf(sgn * (2.0f * PI_F / 16.0f) * k1, &s1, &c1);
    v2f r; r.x = c0; r.y = c1;
    v2f i; i.x = s0; i.y = s1;
    v2f n; n.x = -s0; n.y = -s1;
    w.re[q] = r; w.im[q] = i; w.nim[q] = n;
  }
  return w;
}

// Complex matmul-accumulate: (cr,ci) += W * (Br,Bi), one K=4 chunk chain.
__device__ __forceinline__ void cmat_mac(const W16& w, const v2f Br[4], const v2f Bi[4],
                                         v8f& cr, v8f& ci) {
#pragma unroll
  for (int q = 0; q < 4; ++q) {
    cr = WMMA_F32X4(w.re[q],  Br[q], cr);
    cr = WMMA_F32X4(w.nim[q], Bi[q], cr);   // -Wi * Xi
    ci = WMMA_F32X4(w.re[q],  Bi[q], ci);
    ci = WMMA_F32X4(w.im[q],  Br[q], ci);
  }
}

// ---------------------------------------------------------------------------
// Wave-level 512-pt complex FFT (sgn=-1 fwd, +1 inv; no 1/N scaling).
// in/out: 512 float2 in LDS (may alias). tmp: 16x32 with pitch 33 (bank-safe).
// Decomposition: n = 32*n1 + n2;  k = k1 + 16*(km1 + 16*km2)
//   stage1: T[k1][n2] = sum_n1 W16[k1][n1] * in[32*n1+n2], * W512^{sgn*k1*n2}
//   stage2: per k1, 32-pt DFT over n2 via 16-pt DFT on even/odd + radix-2.
// B-operand layout mirrors A: chunk q, lanes 0-15: v0=K(4q),v1=K(4q+1);
// lanes 16-31: v0=K(4q+2),v1=K(4q+3); column N = lane&15.
// ---------------------------------------------------------------------------
__device__ void fft512(const float2* in, float2* tmp, float2* out,
                       const W16& w, float sgn) {
  const int lane = threadIdx.x & 31;
  const int nloc = lane & 15;
  const int hi = lane >> 4;

  // ---- Stage 1: two 16-column tiles ----
#pragma unroll
  for (int t = 0; t < 2; ++t) {
    v2f Br[4], Bi[4];
#pragma unroll
    for (int q = 0; q < 4; ++q) {
      const int k0 = 4 * q + 2 * hi;                 // n1 for v element 0
      const float2 x0 = in[32 * k0 + 16 * t + nloc];
      const float2 x1 = in[32 * (k0 + 1) + 16 * t + nloc];
      v2f br; br.x = x0.x; br.y = x1.x;
      v2f bi; bi.x = x0.y; bi.y = x1.y;
      Br[q] = br; Bi[q] = bi;
    }
    v8f cr = {}, ci = {};
    cmat_mac(w, Br, Bi, cr, ci);
    // twiddle W512^{sgn*k1*n2}, store T[k1][n2] (pitch 33)
#pragma unroll
    for (int v = 0; v < 8; ++v) {
      const int k1 = v + 8 * hi;
      const int n2 = 16 * t + nloc;
      float sa, ca;
      __sincosf(sgn * (2.0f * PI_F / 512.0f) * (float)(k1 * n2), &sa, &ca);
      const float tr = cr[v] * ca - ci[v] * sa;
      const float ti = cr[v] * sa + ci[v] * ca;
      tmp[k1 * 33 + n2] = make_float2(tr, ti);
    }
  }

  // ---- Stage 2: 32-pt DFT over n2 as 16-pt(even/odd) + radix-2 ----
  v8f u0r = {}, u0i = {}, u1r = {}, u1i = {};
#pragma unroll
  for (int m2 = 0; m2 < 2; ++m2) {
    v2f Br[4], Bi[4];
#pragma unroll
    for (int q = 0; q < 4; ++q) {
      const int m1a = 4 * q + 2 * hi;                 // m1 for v element 0
      const float2 x0 = tmp[nloc * 33 + 2 * m1a + m2];
      const float2 x1 = tmp[nloc * 33 + 2 * (m1a + 1) + m2];
      v2f br; br.x = x0.x; br.y = x1.x;
      v2f bi; bi.x = x0.y; bi.y = x1.y;
      Br[q] = br; Bi[q] = bi;
    }
    v8f cr = {}, ci = {};
    cmat_mac(w, Br, Bi, cr, ci);
    if (m2 == 0) { u0r = cr; u0i = ci; } else { u1r = cr; u1i = ci; }
  }
  // twiddle odd branch by W32^{sgn*km1}, radix-2 combine, scatter out
#pragma unroll
  for (int v = 0; v < 8; ++v) {
    const int km1 = v + 8 * hi;
    float sa, ca;
    __sincosf(sgn * (2.0f * PI_F / 32.0f) * (float)km1, &sa, &ca);
    const float ar = u1r[v] * ca - u1i[v] * sa;
    const float ai = u1r[v] * sa + u1i[v] * ca;
    const int k = nloc + 16 * km1;
    out[k] = make_float2(u0r[v] + ar, u0i[v] + ai);
    out[k + 256] = make_float2(u0r[v] - ar, u0i[v] - ai);
  }
}

// LDS layout for row kernels: rows contiguous (TDM-friendly), then tmp region.
//   row(wave)  = sm + wave*512          (8*512 float2 = 32 KB block)
//   tmp(wave)  = sm + 4096 + wave*528
#define ROW_SHMEM_F2 (8 * 512 + 8 * 528)

// ---------------------------------------------------------------------------
// K1: forward FFT along W. One wave per row; 8 rows per block.
// Results drained to global via async LDS->global stores.
// ---------------------------------------------------------------------------
__global__ __launch_bounds__(256) void k_row_fft_fwd(const float* __restrict__ x,
                                                     float2* __restrict__ F) {
  extern __shared__ float2 sm[];
  const int wave = threadIdx.x >> 5;
  const int lane = threadIdx.x & 31;
  float2* row = sm + wave * 512;
  float2* tmp = sm + 4096 + wave * 528;
  const size_t grow = (size_t)blockIdx.x * 8 + wave;   // 65536 rows total
  const float* src = x + grow * 512;
  __builtin_prefetch(src + lane * 16, 0, 1);           // global_prefetch_b8
#pragma unroll
  for (int j = 0; j < 16; ++j) {
    const int i = lane + 32 * j;
    row[i] = make_float2(src[i], 0.0f);                // real -> complex expand
  }
  const W16 w = make_w16(-1.0f);
  fft512(row, tmp, row, w, -1.0f);
  // Drain: DS writes must land before the async engine reads LDS.
  wait_ds0();
  float2* dst = F + grow * 512;
#pragma unroll
  for (int j = 0; j < 16; ++j) {
    const int i = lane + 32 * j;
    async_l2g_b64(&dst[i], &row[i]);                   // ASYNCcnt path
  }
  // S_ENDPGM waits for ASYNCcnt implicitly.
}

// ---------------------------------------------------------------------------
// K2/K5: FFT along H on 16-column tiles. The transposing gather/scatter between
// global (h,c) and LDS column-major (pitch 513, bank-safe) rides the async-LDS
// engine with per-lane LDS addresses. MASK zeroes the low-frequency square
// (unshifted coords) in LDS before the inverse transform.
// ---------------------------------------------------------------------------
template <bool INVERSE, bool MASK>
__global__ __launch_bounds__(256) void k_col_fft(float2* __restrict__ F,
                                                 const int* __restrict__ cut) {
  extern __shared__ float2 sm[];                 // 16*513 + 8*528 float2
  const int wave = threadIdx.x >> 5;
  float2* colb = sm;
  float2* tmp = sm + 16 * 513 + wave * 528;
  const int img = blockIdx.y;                    // b*16 + c
  const int c0 = blockIdx.x * 16;
  float2* base = F + (size_t)img * (512 * 512) + c0;

  // Async transposing gather: global (h, c0+cc) -> LDS colb[cc*513 + h].
#pragma unroll 4
  for (int idx = threadIdx.x; idx < 512 * 16; idx += 256) {
    const int h = idx >> 4, cc = idx & 15;
    async_g2l_b64(&colb[cc * 513 + h], &base[(size_t)h * 512 + cc]);
  }
  wait_async0();
  __syncthreads();

  if (MASK) {
    const int cb = cut[img >> 4];
    for (int idx = threadIdx.x; idx < 512 * 16; idx += 256) {
      const int h = idx >> 4, cc = idx & 15;
      const int vcol = c0 + cc;
      const int cmh = (h < 256) ? (h + 1) : (512 - h);
      const int cmv = (vcol < 256) ? (vcol + 1) : (512 - vcol);
      const int k = cmh > cmv ? cmh : cmv;
      if (k <= cb) colb[cc * 513 + h] = make_float2(0.0f, 0.0f);
    }
    __syncthreads();
  }

  const float sgn = INVERSE ? 1.0f : -1.0f;
  const W16 w = make_w16(sgn);
#pragma unroll
  for (int s = 0; s < 2; ++s) {
    const int cc = wave * 2 + s;
    fft512(colb + cc * 513, tmp, colb + cc * 513, w, sgn);
  }
  __syncthreads();                               // all DS writes visible

  // Async transposing scatter back to global.
#pragma unroll 4
  for (int idx = threadIdx.x; idx < 512 * 16; idx += 256) {
    const int h = idx >> 4, cc = idx & 15;
    async_l2g_b64(&base[(size_t)h * 512 + cc], &colb[cc * 513 + h]);
  }
  // S_ENDPGM waits for ASYNCcnt implicitly.
}

// ---------------------------------------------------------------------------
// K3: per-batch energy histogram over k = max(cm(u), cm(v)) from channel 8.
// (fftshift over B/C cancels: batch b's cutoff comes from F[b, 8].)
// ---------------------------------------------------------------------------
__global__ __launch_bounds__(256) void k_hist(const float2* __restrict__ F,
                                              float* __restrict__ hist) {
  __shared__ float lh[257];
  for (int i = threadIdx.x; i < 257; i += 256) lh[i] = 0.0f;
  __syncthreads();
  const size_t t = (size_t)blockIdx.x * 256 + threadIdx.x;  // 8*262144 total
  const int b = (int)(t >> 18);
  const int p = (int)(t & 262143);
  const int u = p >> 9, v = p & 511;
  const float2 z = F[((size_t)(b * 16 + 8)) * 262144 + p];
  const float m2 = z.x * z.x + z.y * z.y;
  const int cmu = (u < 256) ? (u + 1) : (512 - u);
  const int cmv = (v < 256) ? (v + 1) : (512 - v);
  const int k = cmu > cmv ? cmu : cmv;
  atomicAdd(&lh[k], m2);
  __syncthreads();
  for (int i = threadIdx.x; i < 257; i += 256)
    if (lh[i] != 0.0f) atomicAdd(&hist[b * 257 + i], lh[i]);
}

__global__ void k_init(float* hist) {
  for (int i = threadIdx.x; i < 8 * 257; i += blockDim.x) hist[i] = 0.0f;
}

// K4: prefix-scan the histogram; first c in [1,255] with cum >= 0.9*total, else 5.
__global__ void k_scan(const float* __restrict__ hist, int* __restrict__ cut) {
  const int b = threadIdx.x;
  if (b >= 8) return;
  float total = 0.0f;
  for (int k = 1; k <= 256; ++k) total += hist[b * 257 + k];
  const float thr = total * 0.9f;
  float run = 0.0f;
  int res = 5;
  bool found = false;
  for (int c = 1; c <= 255; ++c) {
    run += hist[b * 257 + c];
    if (!found && run >= thr) { res = c; found = true; }
  }
  cut[b] = res;
}

// ---------------------------------------------------------------------------
// K6: inverse FFT along W + |z| / (512*512) -> output.
// The block's 8 rows are one contiguous 32 KB chunk -> single TDM descriptor.
// ---------------------------------------------------------------------------
__global__ __launch_bounds__(256) void k_row_ifft_abs(const float2* __restrict__ F,
                                                      float* __restrict__ out) {
  extern __shared__ float2 sm[];
  const int wave = threadIdx.x >> 5;
  const int lane = threadIdx.x & 31;
  float2* row = sm + wave * 512;
  float2* tmp = sm + 4096 + wave * 528;
  const size_t grow = (size_t)blockIdx.x * 8 + wave;

#ifdef USE_TDM
  if (wave == 0) {
    // One DMA: F[blockIdx*4096 .. +4096) float2 -> sm[0..4096)
    tdm_load_1d(lds_off(sm), F + (size_t)blockIdx.x * 4096, 4096u);
    __builtin_amdgcn_s_wait_tensorcnt(0);
  }
  __syncthreads();
#else
  const float2* src = F + grow * 512;
#pragma unroll
  for (int j = 0; j < 16; ++j) {
    const int i = lane + 32 * j;
    row[i] = src[i];
  }
  __syncthreads();
#endif

  const W16 w = make_w16(1.0f);
  fft512(row, tmp, row, w, 1.0f);
  const float scale = 1.0f / (512.0f * 512.0f);
  float* dst = out + grow * 512;
#pragma unroll
  for (int j = 0; j < 16; ++j) {
    const int i = lane + 32 * j;
    const float2 z = row[i];
    dst[i] = sqrtf(z.x * z.x + z.y * z.y) * scale;
  }
}

// ---------------------------------------------------------------------------
extern "C" void kernel_launch(void* const* d_in, const int* in_sizes, int n_in,
                              void* d_out, int out_size, void* d_ws, size_t ws_size,
                              hipStream_t stream) {
  const float* x = (const float*)d_in[0];
  float* out = (float*)d_out;

  // Workspace layout: [complex spectrum 256 MB][hist 8*257 f32][cutoff 8 i32]
  float2* F = (float2*)d_ws;
  float* hist = (float*)((char*)d_ws + (size_t)8 * 16 * 512 * 512 * sizeof(float2));
  int* cut = (int*)(hist + 8 * 257);

  const size_t row_shmem = (size_t)ROW_SHMEM_F2 * sizeof(float2);          // 66,560 B
  const size_t col_shmem = (size_t)(16 * 513 + 8 * 528) * sizeof(float2);  // 99,456 B

  k_init<<<1, 256, 0, stream>>>(hist);
  k_row_fft_fwd<<<8192, 256, row_shmem, stream>>>(x, F);
  k_col_fft<false, false><<<dim3(32, 128), 256, col_shmem, stream>>>(F, nullptr);
  k_hist<<<8192, 256, 0, stream>>>(F, hist);
  k_scan<<<1, 32, 0, stream>>>(hist, cut);
  k_col_fft<true, true><<<dim3(32, 128), 256, col_shmem, stream>>>(F, cut);
  k_row_ifft_abs<<<8192, 256, row_shmem, stream>>>(F, out);
}